// ESN_41807211659336
// MI455X (gfx1250) — compile-verified
//
#include <hip/hip_runtime.h>
#include <math.h>

typedef float v2f __attribute__((ext_vector_type(2)));
typedef float v8f __attribute__((ext_vector_type(8)));
typedef unsigned v4u __attribute__((ext_vector_type(4)));
typedef int v4i __attribute__((ext_vector_type(4)));
typedef int v8i __attribute__((ext_vector_type(8)));

#define RSZ 2048
#define ISZ 32
#define OSZ 32
#define TSZ 8192
#define TSTEPS 8191          // T-1 sequential steps
#define NWG_REC 64
#define ROWS_PER_WG (RSZ / NWG_REC)   // 32 rows of W_res per workgroup

// ---------------------------------------------------------------------------
// Init: zero the grid-barrier state (d_ws is poisoned by the harness).
// ---------------------------------------------------------------------------
__global__ void esn_init(unsigned* __restrict__ bar) {
    if (threadIdx.x < 2) bar[threadIdx.x] = 0u;
}

// ---------------------------------------------------------------------------
// Phase 1: drive[t, r] = sum_i X[t,i] * W_in[r,i]  via v_wmma_f32_16x16x4_f32
// M = t (8192 rows, padded), N = r (2048), K = 32.  One wave per 16x16 tile.
// ---------------------------------------------------------------------------
__global__ void esn_drive_wmma(const float* __restrict__ X,
                               const float* __restrict__ Win,
                               float* __restrict__ S) {
    const int wave  = (blockIdx.x << 3) + (threadIdx.x >> 5);
    const int lane  = threadIdx.x & 31;
    const int mt    = wave >> 7;          // 512 M-tiles
    const int nt    = wave & 127;         // 128 N-tiles
    const int l15   = lane & 15;
    const int khalf = (lane >> 4) << 1;   // lanes 16..31 hold K = k0+2, k0+3
    const int m     = (mt << 4) + l15;    // A row for this lane
    const int n     = (nt << 4) + l15;    // B col for this lane

    v8f acc = {0.f, 0.f, 0.f, 0.f, 0.f, 0.f, 0.f, 0.f};
    #pragma unroll
    for (int k0 = 0; k0 < ISZ; k0 += 4) {
        v2f a, b;
        a.x = X[m * ISZ + k0 + khalf];
        a.y = X[m * ISZ + k0 + khalf + 1];
        b.x = Win[n * ISZ + k0 + khalf];       // B(k, n) = W_in[n, k]
        b.y = Win[n * ISZ + k0 + khalf + 1];
        acc = __builtin_amdgcn_wmma_f32_16x16x4_f32(
            false, a, false, b, (short)0, acc, false, false);
    }
    // C/D layout: VGPR v -> M = v (lanes 0-15) or M = 8+v (lanes 16-31)
    const int rbase = (mt << 4) + ((lane >> 4) << 3);
    const int col   = (nt << 4) + l15;
    #pragma unroll
    for (int v = 0; v < 8; ++v)
        S[(size_t)(rbase + v) * RSZ + col] = acc[v];
}

// ---------------------------------------------------------------------------
// Phase 2: persistent sequential recurrence.
//   s_{t+1} = tanh(S[t] + W_res @ s_t), written in place over S[t].
// 64 WGs x 256 threads; 32 rows per WG; 8 lanes per row.
// State broadcast: TDM tensor_load_to_lds (one instruction moves the 8 KB
// tile, TENSORcnt-tracked), fallback to float4 copies if builtin missing.
// One device-wide generation barrier per step.
// ---------------------------------------------------------------------------
__device__ __forceinline__ void grid_sync(unsigned* __restrict__ bar,
                                          unsigned nwg) {
    __syncthreads();
    if (threadIdx.x == 0) {
        __threadfence();   // publish this WG's state rows device-wide
        unsigned g = __hip_atomic_load(&bar[1], __ATOMIC_RELAXED,
                                       __HIP_MEMORY_SCOPE_AGENT);
        unsigned a = __hip_atomic_fetch_add(&bar[0], 1u, __ATOMIC_ACQ_REL,
                                            __HIP_MEMORY_SCOPE_AGENT);
        if (a == nwg - 1u) {
            __hip_atomic_store(&bar[0], 0u, __ATOMIC_RELAXED,
                               __HIP_MEMORY_SCOPE_AGENT);
            __hip_atomic_store(&bar[1], g + 1u, __ATOMIC_RELEASE,
                               __HIP_MEMORY_SCOPE_AGENT);
        } else {
            while (__hip_atomic_load(&bar[1], __ATOMIC_ACQUIRE,
                                     __HIP_MEMORY_SCOPE_AGENT) == g) {
                __builtin_amdgcn_s_sleep(1);
            }
        }
    }
    __syncthreads();
}

__global__ void esn_recur(const float* __restrict__ Wres,
                          const float* __restrict__ state0,
                          float* __restrict__ S,
                          unsigned* __restrict__ bar) {
    __shared__ float s_lds[RSZ];   // 8 KB broadcast of s_t

    const int tid       = threadIdx.x;
    const int row_local = tid >> 3;                      // 0..31
    const int seg       = tid & 7;                       // 0..7 lanes per row
    const int row       = blockIdx.x * ROWS_PER_WG + row_local;
    const float4* w4    = (const float4*)(Wres + (size_t)row * RSZ);

#if __has_builtin(__builtin_amdgcn_tensor_load_to_lds)
    // LDS byte address of s_lds: generic LDS pointers carry the LDS offset
    // in the low 32 bits (aperture bits live in [63:32]).
    const unsigned lds_base = (unsigned)(size_t)(void*)s_lds;
#endif

    for (int t = 0; t < TSTEPS; ++t) {
        const float* src = (t == 0) ? state0 : (S + (size_t)(t - 1) * RSZ);

#if __has_builtin(__builtin_amdgcn_tensor_load_to_lds)
        // --- TDM: DMA the 2048-float state tile straight into LDS ---------
        if (tid < 32) {                       // one wave issues the descriptor
            const unsigned long long ga = (unsigned long long)(size_t)src;
            v4u g0;
            g0.x = 1u;                                   // count=1, user mode
            g0.y = lds_base;                             // lds_addr
            g0.z = (unsigned)ga;                         // global_addr[31:0]
            g0.w = (unsigned)((ga >> 32) & 0x1FFFFFFull) // global_addr[56:32]
                   | (2u << 30);                         // type=2 ("image")
            v8i g1;
            g1[0] = (int)(2u << 16);      // data_size=4B; wg_mask=0; no pad
            g1[1] = (int)(2048u << 16);   // tensor_dim0[15:0]=2048 @ bits 63:48
            g1[2] = (int)(1u << 16);      // tensor_dim0[31:16]=0; tensor_dim1=1
            g1[3] = (int)(2048u << 16);   // tensor_dim1 hi=0; tile_dim0=2048
            g1[4] = 0;                    // tile_dim1=0 (1D tile), tile_dim2=0
            g1[5] = 2048;                 // tensor_dim0_stride[31:0]
            g1[6] = 0;                    // stride hi / tensor_dim1_stride lo
            g1[7] = 0;
            v4i gz = {0, 0, 0, 0};
#if __has_include(<hip/amd_detail/amd_gfx1250_TDM.h>)
            v8i gz8 = {0, 0, 0, 0, 0, 0, 0, 0};
            __builtin_amdgcn_tensor_load_to_lds(g0, g1, gz, gz, gz8, 0);
#else
            __builtin_amdgcn_tensor_load_to_lds(g0, g1, gz, gz, 0);
#endif
            __builtin_amdgcn_s_wait_tensorcnt(0);
        }
        __syncthreads();
#else
        // --- fallback: cooperative float4 broadcast -----------------------
        {
            const float4* src4 = (const float4*)src;
            float4* l4 = (float4*)s_lds;
            l4[tid]       = src4[tid];
            l4[tid + 256] = src4[tid + 256];
        }
        __syncthreads();
#endif

        // --- dot(W_res[row, :], s_t): seg-interleaved float4, conflict-free
        // element block for (j, seg) = floats [j*32 + seg*4, +4)
        __builtin_prefetch(&w4[seg], 0, 3);   // keep W_res lines warm (L2-resident)
        float acc = 0.f;
        #pragma unroll 8
        for (int j = 0; j < 64; ++j) {
            const int idx = (j << 3) + seg;
            float4 w = w4[idx];
            float4 s = ((const float4*)s_lds)[idx];
            acc = fmaf(w.x, s.x, acc);
            acc = fmaf(w.y, s.y, acc);
            acc = fmaf(w.z, s.z, acc);
            acc = fmaf(w.w, s.w, acc);
        }
        // reduce the 8 adjacent lanes of this row (same wave)
        acc += __shfl_xor(acc, 1, 32);
        acc += __shfl_xor(acc, 2, 32);
        acc += __shfl_xor(acc, 4, 32);

        if (seg == 0) {
            const size_t p = (size_t)t * RSZ + row;
            S[p] = tanhf(S[p] + acc);        // drive + matvec -> new state
        }
        grid_sync(bar, (unsigned)gridDim.x); // step t published before t+1 reads
    }
}

// ---------------------------------------------------------------------------
// Phase 3: preds[t, o] = sum_r S[t,r]*W_out[o,r] + sum_i X[t,i]*W_out[o,R+i]
// M = t (512 tiles), N = o (2 tiles), K = 2080 via f32 WMMA.
// ---------------------------------------------------------------------------
__global__ void esn_readout_wmma(const float* __restrict__ S,
                                 const float* __restrict__ X,
                                 const float* __restrict__ Wout,
                                 float* __restrict__ out) {
    const int wave  = (blockIdx.x << 3) + (threadIdx.x >> 5);  // 1024 waves
    const int lane  = threadIdx.x & 31;
    const int mt    = wave >> 1;          // 512 M-tiles
    const int nt    = wave & 1;           // 2 N-tiles
    const int l15   = lane & 15;
    const int khalf = (lane >> 4) << 1;
    const int m     = (mt << 4) + l15;
    const int n     = (nt << 4) + l15;

    const float* arow = S + (size_t)m * RSZ;
    const float* brow = Wout + (size_t)n * (RSZ + ISZ);

    v8f acc = {0.f, 0.f, 0.f, 0.f, 0.f, 0.f, 0.f, 0.f};
    for (int k0 = 0; k0 < RSZ; k0 += 4) {          // state part of the readout
        v2f a, b;
        a.x = arow[k0 + khalf];
        a.y = arow[k0 + khalf + 1];
        b.x = brow[k0 + khalf];
        b.y = brow[k0 + khalf + 1];
        acc = __builtin_amdgcn_wmma_f32_16x16x4_f32(
            false, a, false, b, (short)0, acc, false, false);
    }
    const float* xrow = X + (size_t)m * ISZ;
    #pragma unroll
    for (int k0 = 0; k0 < ISZ; k0 += 4) {          // input part of the readout
        v2f a, b;
        a.x = xrow[k0 + khalf];
        a.y = xrow[k0 + khalf + 1];
        b.x = brow[RSZ + k0 + khalf];
        b.y = brow[RSZ + k0 + khalf + 1];
        acc = __builtin_amdgcn_wmma_f32_16x16x4_f32(
            false, a, false, b, (short)0, acc, false, false);
    }
    const int rbase = (mt << 4) + ((lane >> 4) << 3);
    const int col   = (nt << 4) + l15;             // 0..31
    #pragma unroll
    for (int v = 0; v < 8; ++v) {
        const int trow = rbase + v;
        if (trow < TSTEPS)                          // drop the padded row 8191
            out[(size_t)trow * OSZ + col] = acc[v];
    }
}

// ---------------------------------------------------------------------------
// Host launcher. Inputs: X[8192,32], W_in[2048,32], W_res[2048,2048],
// W_out[32,2080], state0[2048]. Output: preds[8191,32].
// Workspace: [0,256): barrier; [256, 256+64MB): S (drive, then states in place)
// ---------------------------------------------------------------------------
extern "C" void kernel_launch(void* const* d_in, const int* in_sizes, int n_in,
                              void* d_out, int out_size, void* d_ws, size_t ws_size,
                              hipStream_t stream) {
    const float* X      = (const float*)d_in[0];
    const float* Win    = (const float*)d_in[1];
    const float* Wres   = (const float*)d_in[2];
    const float* Wout   = (const float*)d_in[3];
    const float* state0 = (const float*)d_in[4];
    float* out = (float*)d_out;

    unsigned* bar = (unsigned*)d_ws;
    float* S = (float*)((char*)d_ws + 256);   // 8192 x 2048 f32 = 64 MB

    esn_init<<<1, 64, 0, stream>>>(bar);
    esn_drive_wmma<<<8192, 256, 0, stream>>>(X, Win, S);       // 65536 tiles
    esn_recur<<<NWG_REC, 256, 0, stream>>>(Wres, state0, S, bar);
    esn_readout_wmma<<<128, 256, 0, stream>>>(S, X, Wout, out); // 1024 tiles
    (void)in_sizes; (void)n_in; (void)out_size; (void)ws_size;
}